// EvalMemoryReader_6219112644707
// MI455X (gfx1250) — compile-verified
//
#include <hip/hip_runtime.h>
#include <hip/hip_bf16.h>
#include <math.h>

typedef __attribute__((ext_vector_type(2))) float v2f;
typedef __attribute__((ext_vector_type(8))) float v8f;

#define CK   64
#define CV   512
#define T_   8
#define H_   32
#define W_   56
#define HW   (H_ * W_)     // 1792
#define THW  (T_ * HW)     // 14336
#define TOPK 50
#define INV2KM2 (1.0f / (2.0f * 5.6f * 5.6f))

// ---------- order-preserving float <-> uint encoding ----------
__device__ __forceinline__ unsigned ord_enc(float f) {
    unsigned u = __float_as_uint(f);
    return u ^ ((u & 0x80000000u) ? 0xFFFFFFFFu : 0x80000000u);
}
__device__ __forceinline__ float ord_dec(unsigned u) {
    unsigned b = (u & 0x80000000u) ? (u ^ 0x80000000u) : ~u;
    return __uint_as_float(b);
}
__device__ __forceinline__ unsigned long long shfl_xor_u64(unsigned long long v, int m) {
    unsigned lo = (unsigned)v, hi = (unsigned)(v >> 32);
    lo = __shfl_xor(lo, m, 32);
    hi = __shfl_xor(hi, m, 32);
    return ((unsigned long long)hi << 32) | lo;
}

// ---------- kernel 0: |m|^2, |q|^2 norms + init reduction buffers ----------
__global__ void k0_norms(const float* __restrict__ mk, const float* __restrict__ qk,
                         float* __restrict__ mnorm, float* __restrict__ qnorm,
                         unsigned long long* __restrict__ rowArg,
                         unsigned* __restrict__ colMaxBits) {
    int g = blockIdx.x * blockDim.x + threadIdx.x;
    if (g < THW) {
        float s = 0.f;
        #pragma unroll 8
        for (int c = 0; c < CK; ++c) { float v = mk[(size_t)c * THW + g]; s += v * v; }
        mnorm[g] = s;
        rowArg[g] = 0ull;          // smallest ordered key
    }
    if (g < HW) {
        float s = 0.f;
        #pragma unroll 8
        for (int c = 0; c < CK; ++c) { float v = qk[(size_t)c * HW + g]; s += v * v; }
        qnorm[g] = s;
        colMaxBits[g] = 0u;
    }
}

// ---------- kernel 1: WMMA affinity + fused row-argmax / col-max + coalesced transposed store ----------
// Block = 256 threads = 8 waves; block computes a 128(M) x 16(N) macro-tile, each
// wave one 16x16 WMMA tile, K=64 as 16 chained V_WMMA_F32_16X16X4_F32 steps.
__global__ __launch_bounds__(256) void k1_affinity(
    const float* __restrict__ mk, const float* __restrict__ qk,
    const float* __restrict__ mnorm, const float* __restrict__ qnorm,
    float* __restrict__ affT,
    unsigned long long* __restrict__ rowArg, unsigned* __restrict__ colMaxBits) {
    __shared__ float tile[16][129];    // 129 pad: conflict-free staging
    const int tid  = threadIdx.x;
    const int wave = tid >> 5, lane = tid & 31;
    const int l16 = lane & 15, hi = lane >> 4;
    const int mBlock = blockIdx.x * 128;
    const int qBase  = blockIdx.y * 16;
    const int mTile  = mBlock + wave * 16;
    const int mA = mTile + l16;        // A fragment row (M = lane%16)
    const int qB = qBase + l16;        // B fragment col (N = lane%16)

    v8f acc = {};
    #pragma unroll
    for (int k0 = 0; k0 < CK; k0 += 4) {
        // ISA 32-bit A/B 16x4 layout: lane holds K = k0 + 2*(lane>=16) + {0,1}
        int ka = k0 + hi * 2;
        v2f a, b;
        a.x = mk[(size_t)ka * THW + mA];
        a.y = mk[(size_t)(ka + 1) * THW + mA];
        b.x = qk[(size_t)ka * HW + qB];
        b.y = qk[(size_t)(ka + 1) * HW + qB];
        acc = __builtin_amdgcn_wmma_f32_16x16x4_f32(
            false, a, false, b, (short)0, acc, false, false);
    }

    // epilogue: aff = (2*dot - |m|^2 - |q|^2) / sqrt(CK)
    const float qn = qnorm[qB];
    float aff[8];
    float cmax = -INFINITY;
    #pragma unroll
    for (int e = 0; e < 8; ++e) {
        int m = mTile + e + hi * 8;    // C/D layout: VGPR e -> M = e + 8*(lane>=16)
        float v = (2.0f * acc[e] - mnorm[m] - qn) * 0.125f;
        aff[e] = v;
        cmax = fmaxf(cmax, v);
    }
    // per-query max over memory rows (column max): lanes L and L+16 share q
    cmax = fmaxf(cmax, __shfl_xor(cmax, 16, 32));
    if (hi == 0) atomicMax(&colMaxBits[qB], ord_enc(cmax));

    // per-memory-row argmax over the 16 queries of this tile (lanes 0..15 share row)
    #pragma unroll
    for (int e = 0; e < 8; ++e) {
        unsigned long long key =
            ((unsigned long long)ord_enc(aff[e]) << 32) | (unsigned)qB;
        unsigned long long o;
        o = shfl_xor_u64(key, 1); key = (o > key) ? o : key;
        o = shfl_xor_u64(key, 2); key = (o > key) ? o : key;
        o = shfl_xor_u64(key, 4); key = (o > key) ? o : key;
        o = shfl_xor_u64(key, 8); key = (o > key) ? o : key;
        if (l16 == 0) {
            int m = mTile + e + hi * 8;
            atomicMax(&rowArg[m], key);
        }
    }

    // stage 16q x 128m in LDS, store affT[q][m] coalesced (512B rows)
    #pragma unroll
    for (int e = 0; e < 8; ++e)
        tile[l16][wave * 16 + e + hi * 8] = aff[e];
    __syncthreads();
    for (int i = tid; i < 16 * 128; i += 256) {
        int ql = i >> 7, ml = i & 127;
        affT[(size_t)(qBase + ql) * THW + mBlock + ml] = tile[ql][ml];
    }
}

// ---------- kernel 2: decode argmax centers + per-query max ----------
__global__ void k2_decode(const unsigned long long* __restrict__ rowArg,
                          const unsigned* __restrict__ colMaxBits,
                          float2* __restrict__ centers, float* __restrict__ maxqf) {
    int g = blockIdx.x * blockDim.x + threadIdx.x;
    if (g < THW) {
        unsigned idx = (unsigned)(rowArg[g] & 0xFFFFFFFFull);
        centers[g] = make_float2((float)(idx / W_), (float)(idx % W_));
    }
    if (g < HW) maxqf[g] = ord_dec(colMaxBits[g]);
}

// ---------- kernel 3: per-query top-50 via LDS radix select ----------
// One block per query; keys LDS-resident (57KB of 320KB WGP LDS).
// s = aff - max_q - d^2/(2 km^2); top-k on s == top-k on exp(s)*norm terms.
__global__ __launch_bounds__(512) void k3_topk(
    const float* __restrict__ affT, const float2* __restrict__ centers,
    const float* __restrict__ maxqf,
    int* __restrict__ topIdx, float* __restrict__ topVal) {
    __shared__ unsigned keys[THW];
    __shared__ unsigned hist[256];
    __shared__ unsigned sh_prefix, sh_kRem, sh_cnt, sh_tieCnt;
    __shared__ int   sel[64];
    __shared__ float selw[64];
    __shared__ float sh_sum;

    const int q = blockIdx.x;
    const int tid = threadIdx.x;
    const float maxq = maxqf[q];
    const float yq = (float)(q / W_), xq = (float)(q % W_);

    for (int m = tid; m < THW; m += 512) {
        float aff = affT[(size_t)q * THW + m];
        float2 c = centers[m];
        float dy = yq - c.x, dx = xq - c.y;
        float s = aff - maxq - (dy * dy + dx * dx) * INV2KM2;
        keys[m] = ord_enc(s);
    }
    if (tid == 0) { sh_prefix = 0u; sh_kRem = TOPK; sh_cnt = 0u; sh_tieCnt = 0u; }
    __syncthreads();

    // 4 rounds of 256-bin MSB-first radix select (descending)
    for (int r = 3; r >= 0; --r) {
        if (tid < 256) hist[tid] = 0u;
        __syncthreads();
        unsigned prefix = sh_prefix;
        unsigned hiMask = (r == 3) ? 0u : (0xFFFFFFFFu << ((r + 1) * 8));
        int shift = r * 8;
        for (int m = tid; m < THW; m += 512) {
            unsigned k = keys[m];
            if ((k & hiMask) == prefix) atomicAdd(&hist[(k >> shift) & 255u], 1u);
        }
        __syncthreads();
        if (tid == 0) {
            unsigned kRem = sh_kRem, cum = 0u;
            int b = 255;
            for (; b > 0; --b) {
                unsigned h = hist[b];
                if (cum + h >= kRem) break;
                cum += h;
            }
            sh_prefix = prefix | ((unsigned)b << shift);
            sh_kRem = kRem - cum;
        }
        __syncthreads();
    }
    const unsigned Tkey = sh_prefix;
    const unsigned tiesNeeded = sh_kRem;

    for (int m = tid; m < THW; m += 512) {
        unsigned k = keys[m];
        if (k > Tkey) {
            unsigned p = atomicAdd(&sh_cnt, 1u);
            sel[p] = m;
        } else if (k == Tkey) {
            unsigned t = atomicAdd(&sh_tieCnt, 1u);
            if (t < tiesNeeded) {
                unsigned p = atomicAdd(&sh_cnt, 1u);
                sel[p] = m;
            }
        }
    }
    __syncthreads();
    if (tid < TOPK) selw[tid] = expf(ord_dec(keys[sel[tid]]));
    __syncthreads();
    if (tid == 0) {
        float s = 0.f;
        for (int j = 0; j < TOPK; ++j) s += selw[j];
        sh_sum = s;
    }
    __syncthreads();
    if (tid < TOPK) {
        topIdx[(size_t)q * TOPK + tid] = sel[tid];
        topVal[(size_t)q * TOPK + tid] = selw[tid] / sh_sum;
    }
}

// ---------- generic coalesced transpose (src rows x cols -> dst cols x rows) ----------
__global__ void transpose_k(const float* __restrict__ src, float* __restrict__ dst,
                            int rows, int cols) {
    __shared__ float t[32][33];
    int cBase = blockIdx.x * 32, rBase = blockIdx.y * 32;
    for (int i = threadIdx.y; i < 32; i += 8)
        t[i][threadIdx.x] = src[(size_t)(rBase + i) * cols + cBase + threadIdx.x];
    __syncthreads();
    for (int i = threadIdx.y; i < 32; i += 8)
        dst[(size_t)(cBase + i) * rows + rBase + threadIdx.x] = t[threadIdx.x][i];
}

// ---------- kernel 4: sparse readout, contiguous 2KB gathers from mvT ----------
__global__ __launch_bounds__(512) void k4_readout(
    const float* __restrict__ mvT, const int* __restrict__ topIdx,
    const float* __restrict__ topVal, float* __restrict__ outT) {
    __shared__ int   sm[TOPK];
    __shared__ float sv[TOPK];
    const int q = blockIdx.x, tid = threadIdx.x;
    if (tid < TOPK) {
        sm[tid] = topIdx[(size_t)q * TOPK + tid];
        sv[tid] = topVal[(size_t)q * TOPK + tid];
    }
    __syncthreads();
    float acc = 0.f;
    #pragma unroll 10
    for (int j = 0; j < TOPK; ++j)
        acc = fmaf(sv[j], mvT[(size_t)sm[j] * CV + tid], acc);
    outT[(size_t)q * CV + tid] = acc;
}

extern "C" void kernel_launch(void* const* d_in, const int* in_sizes, int n_in,
                              void* d_out, int out_size, void* d_ws, size_t ws_size,
                              hipStream_t stream) {
    const float* mk = (const float*)d_in[0];
    const float* qk = (const float*)d_in[1];
    const float* mv = (const float*)d_in[2];
    float* out = (float*)d_out;

    char* w = (char*)d_ws;
    auto carve = [&](size_t bytes) -> void* {
        void* p = (void*)w;
        w += (bytes + 255) & ~(size_t)255;
        return p;
    };
    float*              affT       = (float*)carve((size_t)THW * HW * sizeof(float));   // 102.8 MB
    float*              mvT        = (float*)carve((size_t)THW * CV * sizeof(float));   // 29.4 MB
    float*              outT       = (float*)carve((size_t)HW * CV * sizeof(float));    // 3.7 MB
    float*              mnorm      = (float*)carve((size_t)THW * sizeof(float));
    float*              qnorm      = (float*)carve((size_t)HW * sizeof(float));
    unsigned long long* rowArg     = (unsigned long long*)carve((size_t)THW * 8);
    unsigned*           colMaxBits = (unsigned*)carve((size_t)HW * 4);
    float2*             centers    = (float2*)carve((size_t)THW * sizeof(float2));
    float*              maxqf      = (float*)carve((size_t)HW * sizeof(float));
    int*                topIdx     = (int*)carve((size_t)HW * TOPK * sizeof(int));
    float*              topVal     = (float*)carve((size_t)HW * TOPK * sizeof(float));

    k0_norms<<<(THW + 255) / 256, 256, 0, stream>>>(mk, qk, mnorm, qnorm, rowArg, colMaxBits);
    k1_affinity<<<dim3(THW / 128, HW / 16), 256, 0, stream>>>(
        mk, qk, mnorm, qnorm, affT, rowArg, colMaxBits);
    k2_decode<<<(THW + 255) / 256, 256, 0, stream>>>(rowArg, colMaxBits, centers, maxqf);
    k3_topk<<<HW, 512, 0, stream>>>(affT, centers, maxqf, topIdx, topVal);
    transpose_k<<<dim3(THW / 32, CV / 32), dim3(32, 8), 0, stream>>>(mv, mvT, CV, THW);
    k4_readout<<<HW, CV, 0, stream>>>(mvT, topIdx, topVal, outT);
    transpose_k<<<dim3(CV / 32, HW / 32), dim3(32, 8), 0, stream>>>(outT, out, HW, CV);
}